// WHVISquarePow2Matrix_20770461843800
// MI455X (gfx1250) — compile-verified
//
#include <hip/hip_runtime.h>
#include <math.h>

// ---------------------------------------------------------------------------
// WHVI layer, MI455X (gfx1250, wave32).
//   y = s1 ⊙ H( g ⊙ H( s2 ⊙ x ) ),  g = g_mu + softplus(g_rho)*eps
// H4096 = H16 ⊗ H16 ⊗ H16  -> 3 stages of 16x16x16 matmuls per H, done with
// V_WMMA_F32_16X16X4_F32 (4 chained K=4 WMMAs per 16x16 tile), fp32 end to end.
// Memory-bound: 256 MB streamed @ 23.3 TB/s ≈ 11 us; WMMA work ≈ 6.4 GFLOP.
// ---------------------------------------------------------------------------

typedef float v2f __attribute__((ext_vector_type(2)));
typedef float v8f __attribute__((ext_vector_type(8)));

#define DDIM 4096
#define NTOK 8192
#define WAVES_PER_WG 2
#define THREADS (WAVES_PER_WG * 32)
#define NBLOCKS 1024

// Sylvester Hadamard entry: H16[i][j] = (-1)^popcount(i&j)
__device__ __forceinline__ float h16v(int i, int j) {
  return (__popc(i & j) & 1) ? -1.0f : 1.0f;
}

// In-place FWHT of 4096 floats in LDS `buf`, one wave, via WMMA f32 16x16x4.
// If MUL, the final stage's output is multiplied elementwise by lg[idx].
template <bool MUL>
__device__ __forceinline__ void fwht4096_wmma(float* __restrict__ buf,
                                              const float* __restrict__ lg,
                                              int row, int col, int kh,
                                              const v2f hf[4]) {
  // ---- Stage 1: per a-slice, D = V_a(16x16,b x c) * H16   (A=data, B=H)
  for (int a = 0; a < 16; ++a) {
    float* s = buf + (a << 8);
    v8f acc = {0.f, 0.f, 0.f, 0.f, 0.f, 0.f, 0.f, 0.f};
#pragma unroll
    for (int k = 0; k < 4; ++k) {
      const int c0 = 4 * k + 2 * kh;                 // A: K cols c0, c0+1
      v2f A = *(const v2f*)(s + (row << 4) + c0);    // consecutive -> b64 load
      acc = __builtin_amdgcn_wmma_f32_16x16x4_f32(false, A, false, hf[k],
                                                  (short)0, acc, false, false);
    }
#pragma unroll
    for (int r = 0; r < 8; ++r)                      // D: M = r + 8*kh, N = col
      s[((r + (kh << 3)) << 4) + col] = acc[r];
  }
  asm volatile("s_wait_dscnt 0" ::: "memory");

  // ---- Stage 2: per a-slice, D = H16 * W1_a   (A=H, B=data)
  for (int a = 0; a < 16; ++a) {
    float* s = buf + (a << 8);
    v8f acc = {0.f, 0.f, 0.f, 0.f, 0.f, 0.f, 0.f, 0.f};
#pragma unroll
    for (int k = 0; k < 4; ++k) {
      const int r0 = 4 * k + 2 * kh;                 // B: K rows r0, r0+1
      v2f B;
      B[0] = s[(r0 << 4) + col];
      B[1] = s[((r0 + 1) << 4) + col];
      acc = __builtin_amdgcn_wmma_f32_16x16x4_f32(false, hf[k], false, B,
                                                  (short)0, acc, false, false);
    }
#pragma unroll
    for (int r = 0; r < 8; ++r)
      s[((r + (kh << 3)) << 4) + col] = acc[r];
  }
  asm volatile("s_wait_dscnt 0" ::: "memory");

  // ---- Stage 3: per 16-col tile n, transform the a axis: D = H16 * W2[:,n]
  for (int n = 0; n < 16; ++n) {
    v8f acc = {0.f, 0.f, 0.f, 0.f, 0.f, 0.f, 0.f, 0.f};
#pragma unroll
    for (int k = 0; k < 4; ++k) {
      const int a0 = 4 * k + 2 * kh;
      v2f B;
      B[0] = buf[(a0 << 8) + (n << 4) + col];
      B[1] = buf[((a0 + 1) << 8) + (n << 4) + col];
      acc = __builtin_amdgcn_wmma_f32_16x16x4_f32(false, hf[k], false, B,
                                                  (short)0, acc, false, false);
    }
#pragma unroll
    for (int r = 0; r < 8; ++r) {
      const int idx = ((r + (kh << 3)) << 8) + (n << 4) + col;
      float v = acc[r];
      if (MUL) v *= lg[idx];
      buf[idx] = v;
    }
  }
  asm volatile("s_wait_dscnt 0" ::: "memory");
}

__global__ __launch_bounds__(THREADS) void whvi_fwht_wmma_kernel(
    const float* __restrict__ x, const float* __restrict__ s1,
    const float* __restrict__ s2, const float* __restrict__ g_mu,
    const float* __restrict__ g_rho, const float* __restrict__ eps,
    float* __restrict__ out) {
  __shared__ float lbuf[WAVES_PER_WG * DDIM];  // 32 KB: per-wave work buffer
  __shared__ float lg[DDIM];                   // 16 KB: g = mu + softplus(rho)*eps

  for (int i = threadIdx.x; i < DDIM; i += THREADS) {
    const float r = g_rho[i];
    const float sp = (r > 20.0f) ? r : log1pf(__expf(r));
    lg[i] = fmaf(sp, eps[i], g_mu[i]);
  }
  __syncthreads();

  const int lane = threadIdx.x & 31;
  const int warp = threadIdx.x >> 5;
  const int row = lane & 15;   // A/B fragment row (M) for lanes, mod 16
  const int col = lane & 15;   // C/D fragment column (N)
  const int kh  = lane >> 4;   // which K/M half this lane holds
  float* buf = lbuf + warp * DDIM;

  // H16 operand fragment, per K-chunk k (symmetric: serves as A and as B).
  v2f hf[4];
#pragma unroll
  for (int k = 0; k < 4; ++k) {
    const int c0 = 4 * k + 2 * kh;
    v2f t;
    t[0] = h16v(row, c0);
    t[1] = h16v(row, c0 + 1);
    hf[k] = t;
  }

  const int wid = blockIdx.x * WAVES_PER_WG + warp;
  const int nw  = NBLOCKS * WAVES_PER_WG;

  for (int t = wid; t < NTOK; t += nw) {
    const float4* xr  = (const float4*)(x + (size_t)t * DDIM);
    const float4* s24 = (const float4*)s2;

    // Prefetch next token's row (32 lanes * 512 B = 16 KB).
    if (t + nw < NTOK) {
      const char* nxt = (const char*)(x + (size_t)(t + nw) * DDIM);
      __builtin_prefetch(nxt + lane * 512, 0, 1);
    }

    // Load token, fuse *s2, stage into LDS (coalesced b128).
    float4* b4 = (float4*)buf;
    for (int i = lane; i < DDIM / 4; i += 32) {
      float4 v = xr[i];
      const float4 s = s24[i];
      v.x *= s.x; v.y *= s.y; v.z *= s.z; v.w *= s.w;
      b4[i] = v;
    }
    asm volatile("s_wait_dscnt 0" ::: "memory");

    fwht4096_wmma<true>(buf, lg, row, col, kh, hf);   // H, then ⊙ g
    fwht4096_wmma<false>(buf, lg, row, col, kh, hf);  // H

    // Drain LDS, fuse *s1, coalesced b128 store to global.
    float4* orow = (float4*)(out + (size_t)t * DDIM);
    const float4* s14 = (const float4*)s1;
    for (int i = lane; i < DDIM / 4; i += 32) {
      float4 v = b4[i];
      const float4 s = s14[i];
      v.x *= s.x; v.y *= s.y; v.z *= s.z; v.w *= s.w;
      orow[i] = v;
    }
    asm volatile("" ::: "memory");  // keep next iter's LDS writes behind reads
  }
}

extern "C" void kernel_launch(void* const* d_in, const int* in_sizes, int n_in,
                              void* d_out, int out_size, void* d_ws, size_t ws_size,
                              hipStream_t stream) {
  const float* x     = (const float*)d_in[0];
  const float* s1    = (const float*)d_in[1];
  const float* s2    = (const float*)d_in[2];
  const float* g_mu  = (const float*)d_in[3];
  const float* g_rho = (const float*)d_in[4];
  const float* eps   = (const float*)d_in[5];
  // d_in[6] is H: unused — the kernel exploits the Hadamard structure directly.
  float* out = (float*)d_out;

  whvi_fwht_wmma_kernel<<<NBLOCKS, THREADS, 0, stream>>>(x, s1, s2, g_mu, g_rho,
                                                         eps, out);
}